// RegionProposal_69681549410760
// MI455X (gfx1250) — compile-verified
//
#include <hip/hip_runtime.h>
#include <hip/hip_bf16.h>
#include <stdint.h>

// ---------------- problem constants (match reference) ----------------
#define N_ANCH   202500            // 150*150*9
#define S_SORT   262144            // next pow2 >= N_ANCH
#define K_PRE    6000              // PRE_NMS
#define K_PAD    6016              // 188 words * 32 bits
#define W_WORDS  188               // ceil(6000/32)
#define POST     300
#define IOU_THR  0.7f
#define MIN_SZ   (16.0f / 1000.0f)

typedef __attribute__((ext_vector_type(2))) float v2f;
typedef __attribute__((ext_vector_type(8))) float v8f;
typedef unsigned long long u64;
typedef unsigned int       u32;

// ---------------- workspace layout (bytes) ----------------
// roi:   [0, N_ANCH*16)                          = 3,240,000
// keys:  [3,240,000, +S_SORT*8)                  = 2,097,152  -> ends 5,337,152
// btop:  [5,337,152, +K_PAD*16)                  =    96,256  -> ends 5,433,408
// barea: [5,433,408, +K_PAD*4)                   =    24,064  -> ends 5,457,472
// bvalid:[5,457,472, +K_PRE*4)                   =    24,000  -> ends 5,481,472
// keep:  [5,481,472, +K_PRE*4)                   =    24,000  -> ends 5,505,472
// mask:  [0, K_PRE*W_WORDS*4) = 4,512,000  (reuses roi+keys region, dead after gather)
#define ROI_OFF    ((size_t)0)
#define KEYS_OFF   ((size_t)3240000)
#define TOP_OFF    ((size_t)5337152)
#define AREA_OFF   ((size_t)5433408)
#define VALID_OFF  ((size_t)5457472)
#define KEEP_OFF   ((size_t)5481472)
#define MASK_OFF   ((size_t)0)
#define WS_NEEDED  ((size_t)5505472)

__device__ __forceinline__ u32 orderKey(float f) {
    u32 u = __float_as_uint(f);
    return (u & 0x80000000u) ? ~u : (u | 0x80000000u);   // monotonic float->uint
}

// ---------------- 1) softmax score + box decode + clip + key pack ----------------
__global__ __launch_bounds__(256)
void prep_kernel(const float* __restrict__ cls, const float* __restrict__ loc,
                 const float* __restrict__ anc, float4* __restrict__ roi,
                 u64* __restrict__ keys) {
    int i = blockIdx.x * blockDim.x + threadIdx.x;
    if (i >= S_SORT) return;
    if (i >= N_ANCH) { keys[i] = 0ull; return; }          // pad sinks to the end

    __builtin_prefetch(&loc[4 * i], 0, 3);                // global_prefetch_b8

    float c0 = cls[2 * i + 0], c1 = cls[2 * i + 1];
    float s  = 1.0f / (1.0f + expf(c0 - c1));             // softmax[...,1]

    float ax1 = anc[4 * i + 0], ay1 = anc[4 * i + 1];
    float ax2 = anc[4 * i + 2], ay2 = anc[4 * i + 3];
    float pw = ax2 - ax1, ph = ay2 - ay1;
    float pcx = 0.5f * (ax1 + ax2), pcy = 0.5f * (ay1 + ay2);

    float t0 = loc[4 * i + 0], t1 = loc[4 * i + 1];
    float t2 = loc[4 * i + 2], t3 = loc[4 * i + 3];
    float cx = t0 * pw + pcx, cy = t1 * ph + pcy;
    float w = expf(t2) * pw, h = expf(t3) * ph;

    float x1 = fminf(fmaxf(cx - 0.5f * w, 0.0f), 1.0f);
    float y1 = fminf(fmaxf(cy - 0.5f * h, 0.0f), 1.0f);
    float x2 = fminf(fmaxf(cx + 0.5f * w, 0.0f), 1.0f);
    float y2 = fminf(fmaxf(cy + 0.5f * h, 0.0f), 1.0f);

    bool valid = ((x2 - x1) >= MIN_SZ) && ((y2 - y1) >= MIN_SZ);
    float sm = valid ? s : -1.0f;

    roi[i] = make_float4(x1, y1, x2, y2);
    keys[i] = ((u64)orderKey(sm) << 32) | (u64)(0xFFFFFFFFu - (u32)i);
}

// ---------------- 2) bitonic sort, descending (171 passes) ----------------
__global__ __launch_bounds__(256)
void bitonic_kernel(u64* __restrict__ keys, int j, int k) {
    u32 i = blockIdx.x * blockDim.x + threadIdx.x;
    u32 ixj = i ^ (u32)j;
    if (ixj > i) {
        u64 a = keys[i], b = keys[ixj];
        bool upBlock = ((i & (u32)k) == 0u);
        bool doSwap = upBlock ? (a < b) : (a > b);        // overall descending
        if (doSwap) { keys[i] = b; keys[ixj] = a; }
    }
}

// ---------------- 3) gather top-6000 boxes / areas / valid ----------------
__global__ __launch_bounds__(256)
void gather_kernel(const u64* __restrict__ keys, const float4* __restrict__ roi,
                   float4* __restrict__ btop, float* __restrict__ barea,
                   int* __restrict__ bvalid) {
    int t = blockIdx.x * blockDim.x + threadIdx.x;
    if (t >= K_PAD) return;
    if (t < K_PRE) {
        u64 key = keys[t];
        u32 hi  = (u32)(key >> 32);
        u32 idx = 0xFFFFFFFFu - (u32)(key & 0xFFFFFFFFull);
        if (idx >= N_ANCH) idx = N_ANCH - 1;              // safety clamp
        float4 b = roi[idx];
        btop[t]  = b;
        barea[t] = (b.z - b.x) * (b.w - b.y);
        bvalid[t] = (hi >= 0x80000000u) ? 1 : 0;          // score >= 0.0
    } else {
        btop[t]  = make_float4(0.f, 0.f, 0.f, 0.f);       // pad: zero box/area
        barea[t] = 0.f;
    }
}

// ---------------- 4) suppression-mask build: WMMA area-sums + wave32 ballot ----
// One wave per block covers a 16-row x 32-col tile (= one mask word per row).
// D[m][n] = area_m + area_n computed by V_WMMA_F32_16X16X4_F32 with
// A[m][:] = {area_m, 1, 0, 0},  B[:][n] = {1, area_n, area_n, 0}^T  (layout-robust fill).
__global__ __launch_bounds__(32)
void mask_kernel(const float4* __restrict__ btop, const float* __restrict__ barea,
                 u32* __restrict__ mask) {
    const int l    = threadIdx.x;         // 0..31, one wave
    const int wb   = blockIdx.x;          // word index 0..187
    const int r0   = blockIdx.y * 16;     // row tile base
    const int c0   = wb * 32;             // col base (two 16-wide WMMA tiles)
    const int half = l >> 4;
    const int n    = l & 15;

    // A operand (16x4): lanes 0-15 carry M=l -> {K0=area_row, K1=1}; lanes 16-31 -> {K2,K3}={0,0}
    v2f a;
    a.x = (l < 16) ? barea[r0 + n] : 0.0f;
    a.y = (l < 16) ? 1.0f          : 0.0f;

    // B operands (4x16), robust to either K<->(VGPR,lane-half) assignment:
    // lanes 0-15 vgpr0 = K0 = 1;  ambiguous slots both get area_col (K1=K2=area, A1=1,A2=0); K3 = 0.
    float caL = barea[c0 + n];
    float caR = barea[c0 + 16 + n];
    v2f bL, bR;
    bL.x = (l < 16) ? 1.0f : caL;   bL.y = (l < 16) ? caL : 0.0f;
    bR.x = (l < 16) ? 1.0f : caR;   bR.y = (l < 16) ? caR : 0.0f;

    v8f zero = {};
    v8f dL = __builtin_amdgcn_wmma_f32_16x16x4_f32(false, a, false, bL, (short)0, zero, false, false);
    v8f dR = __builtin_amdgcn_wmma_f32_16x16x4_f32(false, a, false, bR, (short)0, zero, false, false);

    float4 cbL = btop[c0 + n];
    float4 cbR = btop[c0 + 16 + n];
    float4 rb[8];
#pragma unroll
    for (int v = 0; v < 8; ++v) rb[v] = btop[r0 + half * 8 + v];

#pragma unroll
    for (int v = 0; v < 8; ++v) {
        const int row = r0 + half * 8 + v;                // matches C/D lane layout
        // left 16-col tile
        float w0 = fminf(rb[v].z, cbL.z) - fmaxf(rb[v].x, cbL.x);
        float h0 = fminf(rb[v].w, cbL.w) - fmaxf(rb[v].y, cbL.y);
        float interL = fmaxf(w0, 0.f) * fmaxf(h0, 0.f);
        float iouL = interL / (dL[v] - interL + 1e-12f);  // dL = area_row + area_col
        bool pL = (iouL > IOU_THR) && ((c0 + n) > row);
        u32 ballotL = __builtin_amdgcn_ballot_w32(pL);
        // right 16-col tile
        float w1 = fminf(rb[v].z, cbR.z) - fmaxf(rb[v].x, cbR.x);
        float h1 = fminf(rb[v].w, cbR.w) - fmaxf(rb[v].y, cbR.y);
        float interR = fmaxf(w1, 0.f) * fmaxf(h1, 0.f);
        float iouR = interR / (dR[v] - interR + 1e-12f);
        bool pR = (iouR > IOU_THR) && ((c0 + 16 + n) > row);
        u32 ballotR = __builtin_amdgcn_ballot_w32(pR);

        if (l == 0) {
            // ballot bits [15:0] belong to row v, bits [31:16] to row v+8 (same 16 cols)
            mask[(size_t)(r0 + v) * W_WORDS + wb] =
                (ballotL & 0xFFFFu) | ((ballotR & 0xFFFFu) << 16);
            mask[(size_t)(r0 + v + 8) * W_WORDS + wb] =
                (ballotL >> 16) | (ballotR & 0xFFFF0000u);
        }
    }
}

// ---------------- 5) serial greedy NMS scan (one block) ----------------
__global__ __launch_bounds__(256)
void nms_scan_kernel(const u32* __restrict__ mask, const int* __restrict__ bvalid,
                     int* __restrict__ keep) {
    __shared__ u32 remv[W_WORDS];
    __shared__ int active;
    int t = threadIdx.x;
    if (t < W_WORDS) remv[t] = 0u;
    __syncthreads();
    for (int i = 0; i < K_PRE; ++i) {
        if (t == 0) {
            int suppressed = (remv[i >> 5] >> (i & 31)) & 1;
            active = (!suppressed) && bvalid[i];
            keep[i] = active;
        }
        __syncthreads();
        if (active && t < W_WORDS) remv[t] |= mask[(size_t)i * W_WORDS + t];
        __syncthreads();
    }
}

// ---------------- 6) compact kept boxes into 300 output rows ----------------
__global__ __launch_bounds__(256)
void finalize_kernel(const int* __restrict__ keep, const float4* __restrict__ btop,
                     float4* __restrict__ out) {
    __shared__ int sel[POST];
    __shared__ int cnt;
    if (threadIdx.x == 0) {
        int c = 0;
        for (int i = 0; i < K_PRE && c < POST; ++i)
            if (keep[i]) sel[c++] = i;
        cnt = c;
    }
    __syncthreads();
    for (int r = threadIdx.x; r < POST; r += blockDim.x)
        out[r] = (r < cnt) ? btop[sel[r]] : make_float4(0.f, 0.f, 0.f, 0.f);
}

// ---------------- host side ----------------
extern "C" void kernel_launch(void* const* d_in, const int* in_sizes, int n_in,
                              void* d_out, int out_size, void* d_ws, size_t ws_size,
                              hipStream_t stream) {
    (void)in_sizes; (void)n_in; (void)out_size;
    if (ws_size < WS_NEEDED) return;   // deterministic no-op if scratch too small

    const float* cls = (const float*)d_in[0];   // (1, N, 2) f32
    const float* loc = (const float*)d_in[1];   // (1, N, 4) f32
    const float* anc = (const float*)d_in[2];   // (N, 4)    f32

    char* ws = (char*)d_ws;
    float4* roi    = (float4*)(ws + ROI_OFF);
    u64*    keys   = (u64*)  (ws + KEYS_OFF);
    float4* btop   = (float4*)(ws + TOP_OFF);
    float*  barea  = (float*)(ws + AREA_OFF);
    int*    bvalid = (int*)  (ws + VALID_OFF);
    int*    keep   = (int*)  (ws + KEEP_OFF);
    u32*    mask   = (u32*)  (ws + MASK_OFF);   // reuses roi/keys region after gather

    prep_kernel<<<S_SORT / 256, 256, 0, stream>>>(cls, loc, anc, roi, keys);

    for (int k = 2; k <= S_SORT; k <<= 1)
        for (int j = k >> 1; j >= 1; j >>= 1)
            bitonic_kernel<<<S_SORT / 256, 256, 0, stream>>>(keys, j, k);

    gather_kernel<<<(K_PAD + 255) / 256, 256, 0, stream>>>(keys, roi, btop, barea, bvalid);

    dim3 mgrid(W_WORDS, K_PRE / 16);            // 188 x 375 tiles, one wave each
    mask_kernel<<<mgrid, 32, 0, stream>>>(btop, barea, mask);

    nms_scan_kernel<<<1, 256, 0, stream>>>(mask, bvalid, keep);
    finalize_kernel<<<1, 256, 0, stream>>>(keep, btop, (float4*)d_out);
}